// ExtendedGCN_34445637714075
// MI455X (gfx1250) — compile-verified
//
#include <hip/hip_runtime.h>
#include <math.h>

typedef __attribute__((ext_vector_type(2))) float v2f;
typedef __attribute__((ext_vector_type(8))) float v8f;

// ---------------------------------------------------------------------------
// Dense GEMM via V_WMMA_F32_16X16X4_F32 (exact fp32 matrix pipe).
// One wave per 16-row tile; the wave produces 16 x DOUT, reusing the A
// fragment across DOUT/16 N-tiles. X:[n,din] row-major, W:[din,DOUT] row-major.
// ---------------------------------------------------------------------------
template <int DOUT>
__global__ __launch_bounds__(32) void gcn_gemm_wmma_f32(
    const float* __restrict__ X, const float* __restrict__ W,
    float* __restrict__ H, int n, int din) {
  constexpr int NT = DOUT / 16;
  const int m0 = blockIdx.x * 16;
  const int lane = threadIdx.x;   // 0..31, wave32
  const int half = lane >> 4;     // 0 or 1
  const int lid  = lane & 15;

  v8f acc[NT];
#pragma unroll
  for (int t = 0; t < NT; ++t)
#pragma unroll
    for (int r = 0; r < 8; ++r) acc[t][r] = 0.0f;

  int mrow = m0 + lid;
  if (mrow >= n) mrow = n - 1;  // clamp (harmless; tail rows never stored)
  const float* xrow = X + (size_t)mrow * din + 2 * half;

  for (int k = 0; k < din; k += 4) {
    // A fragment: lane holds A[m][kb], A[m][kb+1], kb = k + 2*half
    v2f a;
    a.x = xrow[k];
    a.y = xrow[k + 1];
    const int kb = k + 2 * half;
    const float* w0 = W + (size_t)kb * DOUT + lid;
#pragma unroll
    for (int t = 0; t < NT; ++t) {
      // B fragment: lane holds B[kb][n0+lid], B[kb+1][n0+lid]
      v2f b;
      b.x = w0[t * 16];
      b.y = w0[t * 16 + DOUT];
      acc[t] = __builtin_amdgcn_wmma_f32_16x16x4_f32(
          false, a, false, b, (short)0, acc[t], false, false);
    }
  }

  // C/D layout: VGPR r -> M = r + 8*half, N = lid (ISA 7.12.2)
#pragma unroll
  for (int t = 0; t < NT; ++t) {
#pragma unroll
    for (int r = 0; r < 8; ++r) {
      const int mm = m0 + r + 8 * half;
      if (mm < n) H[(size_t)mm * DOUT + t * 16 + lid] = acc[t][r];
    }
  }
}

// ---------------------------------------------------------------------------
// Degree / normalization
// ---------------------------------------------------------------------------
__global__ void zero_i32(int* p, int n) {
  int i = blockIdx.x * blockDim.x + threadIdx.x;
  if (i < n) p[i] = 0;
}

__global__ void count_deg(const int* __restrict__ dst, int* __restrict__ deg, int E) {
  int e = blockIdx.x * blockDim.x + threadIdx.x;
  if (e < E) atomicAdd(&deg[dst[e]], 1);
}

__global__ void make_dinv(const int* __restrict__ deg, float* __restrict__ dinv, int n) {
  int i = blockIdx.x * blockDim.x + threadIdx.x;
  if (i < n) dinv[i] = rsqrtf((float)(deg[i] + 1));  // +1 self loop; deg>=1
}

// ---------------------------------------------------------------------------
// Aggregation: out[i,:] = dinv[i]^2 * h[i,:] + b  (self loop + bias), then
// edge scatter adds dinv[s]*dinv[d] * h[s,:] into row d with f32 atomics.
// ---------------------------------------------------------------------------
__global__ void agg_init(const float* __restrict__ h, const float* __restrict__ dinv,
                         const float* __restrict__ bias, float* __restrict__ out,
                         int n, int dout) {
  int idx = blockIdx.x * blockDim.x + threadIdx.x;
  if (idx >= n * dout) return;
  int i = idx / dout, f = idx - i * dout;
  float di = dinv[i];
  out[idx] = di * di * h[idx] + bias[f];
}

template <int DOUT>
__global__ void edge_scatter(const float* __restrict__ h, const float* __restrict__ dinv,
                             const int* __restrict__ src, const int* __restrict__ dst,
                             float* __restrict__ out, int E) {
  constexpr int CH = DOUT / 4;  // float4 chunks per row
  long long idx = (long long)blockIdx.x * blockDim.x + threadIdx.x;
  if (idx >= (long long)E * CH) return;
  int e = (int)(idx / CH);
  int q = (int)(idx - (long long)e * CH);
  int s = src[e], d = dst[e];
  float c = dinv[s] * dinv[d];
  const float4 v = *(const float4*)(h + (size_t)s * DOUT + q * 4);
  float* op = out + (size_t)d * DOUT + q * 4;
  atomicAdd(op + 0, c * v.x);
  atomicAdd(op + 1, c * v.y);
  atomicAdd(op + 2, c * v.z);
  atomicAdd(op + 3, c * v.w);
}

__global__ void relu_inplace(float* p, int n) {
  int i = blockIdx.x * blockDim.x + threadIdx.x;
  if (i < n) p[i] = fmaxf(p[i], 0.0f);
}

__global__ void softmax16(const float* __restrict__ logits, float* __restrict__ out, int n) {
  int i = blockIdx.x * blockDim.x + threadIdx.x;
  if (i >= n) return;
  const float* r = logits + (size_t)i * 16;
  float m = r[0];
#pragma unroll
  for (int f = 1; f < 16; ++f) m = fmaxf(m, r[f]);
  float e[16], s = 0.0f;
#pragma unroll
  for (int f = 0; f < 16; ++f) { e[f] = expf(r[f] - m); s += e[f]; }
  float inv = 1.0f / s;
  float* o = out + (size_t)i * 16;
#pragma unroll
  for (int f = 0; f < 16; ++f) o[f] = e[f] * inv;
}

// ---------------------------------------------------------------------------
// Orchestration
// ---------------------------------------------------------------------------
extern "C" void kernel_launch(void* const* d_in, const int* in_sizes, int n_in,
                              void* d_out, int out_size, void* d_ws, size_t ws_size,
                              hipStream_t stream) {
  const float* x  = (const float*)d_in[0];
  const int* ei   = (const int*)d_in[1];
  const float* W1 = (const float*)d_in[2];
  const float* b1 = (const float*)d_in[3];
  const float* W2 = (const float*)d_in[4];
  const float* b2 = (const float*)d_in[5];
  const float* W3 = (const float*)d_in[6];
  const float* b3 = (const float*)d_in[7];

  const int n = in_sizes[0] / 128;
  const int E = in_sizes[1] / 2;
  const int* src = ei;
  const int* dst = ei + E;

  float* hbuf = (float*)d_ws;                 // n*128
  float* abuf = hbuf + (size_t)n * 128;       // n*128
  float* dinv = abuf + (size_t)n * 128;       // n
  int*   degi = (int*)(dinv + n);             // n

  const int B = 256;
  auto blk = [](long long t, int b) { return (unsigned)((t + b - 1) / b); };

  // degrees -> dinv
  zero_i32<<<blk(n, B), B, 0, stream>>>(degi, n);
  count_deg<<<blk(E, B), B, 0, stream>>>(dst, degi, E);
  make_dinv<<<blk(n, B), B, 0, stream>>>(degi, dinv, n);

  const unsigned mt = blk(n, 16);  // 16-row M tiles

  // ---- layer 1: [n,128]x[128,128], aggregate, relu
  gcn_gemm_wmma_f32<128><<<mt, 32, 0, stream>>>(x, W1, hbuf, n, 128);
  agg_init<<<blk((long long)n * 128, B), B, 0, stream>>>(hbuf, dinv, b1, abuf, n, 128);
  edge_scatter<128><<<blk((long long)E * 32, B), B, 0, stream>>>(hbuf, dinv, src, dst, abuf, E);
  relu_inplace<<<blk((long long)n * 128, B), B, 0, stream>>>(abuf, n * 128);

  // ---- layer 2: [n,128]x[128,64], aggregate
  gcn_gemm_wmma_f32<64><<<mt, 32, 0, stream>>>(abuf, W2, hbuf, n, 128);
  agg_init<<<blk((long long)n * 64, B), B, 0, stream>>>(hbuf, dinv, b2, abuf, n, 64);
  edge_scatter<64><<<blk((long long)E * 16, B), B, 0, stream>>>(hbuf, dinv, src, dst, abuf, E);

  // ---- layer 3: [n,64]x[64,16], aggregate
  gcn_gemm_wmma_f32<16><<<mt, 32, 0, stream>>>(abuf, W3, hbuf, n, 64);
  agg_init<<<blk((long long)n * 16, B), B, 0, stream>>>(hbuf, dinv, b3, abuf, n, 16);
  edge_scatter<16><<<blk((long long)E * 4, B), B, 0, stream>>>(hbuf, dinv, src, dst, abuf, E);

  // softmax over 16 classes
  softmax16<<<blk(n, B), B, 0, stream>>>(abuf, (float*)d_out, n);
}